// NTXent_Loss_30648886624858
// MI455X (gfx1250) — compile-verified
//
#include <hip/hip_runtime.h>
#include <math.h>

typedef __attribute__((ext_vector_type(2))) float v2f;
typedef __attribute__((ext_vector_type(8))) float v8f;

#define B_ROWS 4096
#define D_DIM  256
#define N_ROWS 8192          // 2B
#define LDS_STRIDE 260       // 256 + 4 floats pad -> conflict-free ds_load_b64
#define EXP2C 7.389056098930650f   // exp(2.0) = exp(1/TEMP)

// ---------------------------------------------------------------------------
// Kernel 1: row-normalize z = cat(z1,z2); also zero the denom accumulator.
// One wave (32 lanes) per row; each lane owns 8 consecutive floats.
// ---------------------------------------------------------------------------
__global__ __launch_bounds__(256)
void ntx_normalize(const float* __restrict__ z1, const float* __restrict__ z2,
                   float* __restrict__ zn, float* __restrict__ denom) {
    const int wid  = (blockIdx.x * blockDim.x + threadIdx.x) >> 5;   // row
    const int lane = threadIdx.x & 31;
    if (wid >= N_ROWS) return;

    const float* src = (wid < B_ROWS) ? (z1 + (size_t)wid * D_DIM)
                                      : (z2 + (size_t)(wid - B_ROWS) * D_DIM);
    const float4* s4 = (const float4*)src;
    float4 a = s4[lane * 2];
    float4 b = s4[lane * 2 + 1];

    float ss = a.x*a.x + a.y*a.y + a.z*a.z + a.w*a.w
             + b.x*b.x + b.y*b.y + b.z*b.z + b.w*b.w;
    #pragma unroll
    for (int m = 1; m < 32; m <<= 1) ss += __shfl_xor(ss, m, 32);

    const float r = 1.0f / fmaxf(sqrtf(ss), 1e-8f);

    float4 oa, ob;
    oa.x = a.x*r; oa.y = a.y*r; oa.z = a.z*r; oa.w = a.w*r;
    ob.x = b.x*r; ob.y = b.y*r; ob.z = b.z*r; ob.w = b.w*r;

    float4* d4 = (float4*)(zn + (size_t)wid * D_DIM);
    d4[lane * 2]     = oa;
    d4[lane * 2 + 1] = ob;

    if (lane == 0) denom[wid] = 0.0f;        // zero accumulator for kernel 2
}

// ---------------------------------------------------------------------------
// Kernel 2: fused sim-tile + exp-rowsum.
// Block = 16 rows of zn (staged in LDS). 8 waves; each wave computes 4
// 16x16 output tiles at a time via V_WMMA_F32_16X16X4_F32, K = 256.
// exp(2*sim) is reduced per row and atomically added into denom.
// ---------------------------------------------------------------------------
__global__ __launch_bounds__(256)
void ntx_sim(const float* __restrict__ zn, float* __restrict__ denom) {
    __shared__ float lds[16 * LDS_STRIDE];

    const int tid  = threadIdx.x;
    const int row0 = blockIdx.x * 16;

    // Stage the 16x256 A-tile: 256 threads, 16 floats (4 x float4) each.
    {
        const int r = tid >> 4;              // 0..15 row
        const int c = (tid & 15) * 16;       // 0..240 col
        const float4* src = (const float4*)(zn + (size_t)(row0 + r) * D_DIM + c);
        float4* dst = (float4*)(lds + r * LDS_STRIDE + c);   // 16B aligned (1040 % 16 == 0)
        dst[0] = src[0]; dst[1] = src[1]; dst[2] = src[2]; dst[3] = src[3];
    }
    __syncthreads();

    const int wave = tid >> 5;
    const int lane = tid & 31;
    const int m    = lane & 15;
    const int hi   = lane >> 4;              // 0: K=(k,k+1), 1: K=(k+2,k+3)

    const float* aptr = lds + m * LDS_STRIDE + 2 * hi;

    float racc[8];
    #pragma unroll
    for (int r = 0; r < 8; ++r) racc[r] = 0.0f;

    // 512 column tiles; 8 waves x 4 tiles each -> 16 outer iterations.
    for (int jt = wave * 4; jt < 512; jt += 32) {
        const float* b0 = zn + (size_t)(jt * 16 + m) * D_DIM + 2 * hi;
        const float* b1 = b0 + 16 * D_DIM;
        const float* b2 = b1 + 16 * D_DIM;
        const float* b3 = b2 + 16 * D_DIM;

        v8f c0 = {}, c1 = {}, c2 = {}, c3 = {};

        #pragma unroll 8
        for (int k = 0; k < D_DIM; k += 4) {
            v2f a   = *(const v2f*)(aptr + k);     // ds_load_b64, conflict-free
            v2f vb0 = *(const v2f*)(b0 + k);       // global_load_b64, L2-resident
            v2f vb1 = *(const v2f*)(b1 + k);
            v2f vb2 = *(const v2f*)(b2 + k);
            v2f vb3 = *(const v2f*)(b3 + k);
            c0 = __builtin_amdgcn_wmma_f32_16x16x4_f32(false, a, false, vb0, (short)0, c0, false, false);
            c1 = __builtin_amdgcn_wmma_f32_16x16x4_f32(false, a, false, vb1, (short)0, c1, false, false);
            c2 = __builtin_amdgcn_wmma_f32_16x16x4_f32(false, a, false, vb2, (short)0, c2, false, false);
            c3 = __builtin_amdgcn_wmma_f32_16x16x4_f32(false, a, false, vb3, (short)0, c3, false, false);
        }

        // D layout: VGPR r holds row (r + 8*hi), col = lane&15 within tile.
        #pragma unroll
        for (int r = 0; r < 8; ++r) {
            racc[r] += __expf(2.0f * c0[r]) + __expf(2.0f * c1[r])
                     + __expf(2.0f * c2[r]) + __expf(2.0f * c3[r]);
        }
    }

    // Reduce over the 16 columns held by each half-wave, then atomically
    // accumulate into the global per-row denominator.
    #pragma unroll
    for (int r = 0; r < 8; ++r) {
        float v = racc[r];
        v += __shfl_xor(v, 1, 32);
        v += __shfl_xor(v, 2, 32);
        v += __shfl_xor(v, 4, 32);
        v += __shfl_xor(v, 8, 32);
        if (m == 0) atomicAdd(&denom[row0 + r + 8 * hi], v);
    }
}

// ---------------------------------------------------------------------------
// Kernel 3: positive-pair dots  pos[i] = zn[i] . zn[i+B], one wave per i.
// ---------------------------------------------------------------------------
__global__ __launch_bounds__(256)
void ntx_pos(const float* __restrict__ zn, float* __restrict__ posdot) {
    const int wid  = (blockIdx.x * blockDim.x + threadIdx.x) >> 5;
    const int lane = threadIdx.x & 31;
    if (wid >= B_ROWS) return;

    const float4* p = (const float4*)(zn + (size_t)wid * D_DIM);
    const float4* q = (const float4*)(zn + (size_t)(wid + B_ROWS) * D_DIM);
    float4 a0 = p[lane * 2], a1 = p[lane * 2 + 1];
    float4 b0 = q[lane * 2], b1 = q[lane * 2 + 1];

    float s = a0.x*b0.x + a0.y*b0.y + a0.z*b0.z + a0.w*b0.w
            + a1.x*b1.x + a1.y*b1.y + a1.z*b1.z + a1.w*b1.w;
    #pragma unroll
    for (int mI = 1; mI < 32; mI <<= 1) s += __shfl_xor(s, mI, 32);

    if (lane == 0) posdot[wid] = s;
}

// ---------------------------------------------------------------------------
// Kernel 4: loss = sum_i [ log(denom_i - e^2) - 2*pos_{i mod B} ] * (B/2)
// ---------------------------------------------------------------------------
__global__ __launch_bounds__(256)
void ntx_loss(const float* __restrict__ denom, const float* __restrict__ posdot,
              float* __restrict__ out) {
    __shared__ float sm[256];
    float s = 0.0f;
    for (int i = threadIdx.x; i < N_ROWS; i += 256) {
        const float d = denom[i] - EXP2C;                 // subtract self-sim exp(1/T)
        const float p = posdot[i & (B_ROWS - 1)];
        s += logf(d) - 2.0f * p;                          // -log(exp(2p)/d)
    }
    sm[threadIdx.x] = s;
    __syncthreads();
    for (int k = 128; k > 0; k >>= 1) {
        if (threadIdx.x < k) sm[threadIdx.x] += sm[threadIdx.x + k];
        __syncthreads();
    }
    if (threadIdx.x == 0) out[0] = sm[0] * (0.5f * (float)B_ROWS);  // sum/2*B
}

// ---------------------------------------------------------------------------
extern "C" void kernel_launch(void* const* d_in, const int* in_sizes, int n_in,
                              void* d_out, int out_size, void* d_ws, size_t ws_size,
                              hipStream_t stream) {
    const float* z1 = (const float*)d_in[0];
    const float* z2 = (const float*)d_in[1];
    float* out = (float*)d_out;

    char* ws = (char*)d_ws;
    float* zn     = (float*)ws;                                    // 8 MB
    float* denom  = (float*)(ws + (size_t)N_ROWS * D_DIM * sizeof(float));
    float* posdot = denom + N_ROWS;

    ntx_normalize<<<N_ROWS / 8, 256, 0, stream>>>(z1, z2, zn, denom);
    ntx_sim<<<512, 256, 0, stream>>>(zn, denom);
    ntx_pos<<<B_ROWS / 8, 256, 0, stream>>>(zn, posdot);
    ntx_loss<<<1, 256, 0, stream>>>(denom, posdot, out);
}